// GAT_22565758173841
// MI455X (gfx1250) — compile-verified
//
#include <hip/hip_runtime.h>
#include <hip/hip_bf16.h>
#include <math.h>

typedef float v2f __attribute__((ext_vector_type(2)));
typedef float v8f __attribute__((ext_vector_type(8)));

#define HEADS 4
#define NEG_SLOPE 0.2f

// ---------------------------------------------------------------------------
// GEMM via V_WMMA_F32_16X16X4_F32.
// A: [n, fin] row-major (node features), W: [fin, ncol] row-major.
// Block = 128 threads (4 waves). Each wave owns a 16x64 output strip
// (4 accumulators) so one A fragment feeds 4 independent WMMAs per K-step.
// Block covers 16 rows x 256 cols; grid = (n/16, ncol/256).
// Fragment layouts per ISA 7.12.2:
//   A (32-bit 16x4): lanes 0-15 hold K=kk,kk+1 of row M=lane;
//                    lanes 16-31 hold K=kk+2,kk+3 of row M=lane-16.
//   B mirrors with M<->N.  D: VGPR r -> (M=r, N=lane) lanes 0-15,
//                                      (M=r+8, N=lane-16) lanes 16-31.
// ---------------------------------------------------------------------------
__global__ void gemm_wmma_f32(const float* __restrict__ A,
                              const float* __restrict__ W,
                              float* __restrict__ C,
                              int fin, int ncol) {
  __shared__ float As[16 * 256];
  const int rowBase = blockIdx.x * 16;
  const int wave    = threadIdx.x >> 5;
  const int lane    = threadIdx.x & 31;
  const int half    = lane >> 4;      // 0: K pair (0,1); 1: K pair (2,3)
  const int l15     = lane & 15;
  const int colBase = blockIdx.y * 256 + wave * 64;

  // Cooperative, coalesced load of the 16 x fin A tile into LDS.
  for (int i = threadIdx.x; i < 16 * fin; i += blockDim.x) {
    int r = i / fin, c = i - r * fin;
    As[i] = A[(size_t)(rowBase + r) * fin + c];
  }
  __syncthreads();

  v8f acc0 = {}, acc1 = {}, acc2 = {}, acc3 = {};
  for (int kk = 0; kk < fin; kk += 4) {
    const int ka = kk + half * 2;
    v2f a;
    a.x = As[l15 * fin + ka];
    a.y = As[l15 * fin + ka + 1];
    const float* w0 = W + (size_t)ka * ncol + colBase + l15;
    const float* w1 = w0 + ncol;
    // Prefetch next K-panel of W into GL2 (global_prefetch_b8).
    if (kk + 4 < fin) __builtin_prefetch(w0 + 4 * (size_t)ncol, 0, 1);
    v2f b0, b1, b2, b3;
    b0.x = w0[0];   b0.y = w1[0];
    b1.x = w0[16];  b1.y = w1[16];
    b2.x = w0[32];  b2.y = w1[32];
    b3.x = w0[48];  b3.y = w1[48];
    acc0 = __builtin_amdgcn_wmma_f32_16x16x4_f32(false, a, false, b0, (short)0, acc0, false, false);
    acc1 = __builtin_amdgcn_wmma_f32_16x16x4_f32(false, a, false, b1, (short)0, acc1, false, false);
    acc2 = __builtin_amdgcn_wmma_f32_16x16x4_f32(false, a, false, b2, (short)0, acc2, false, false);
    acc3 = __builtin_amdgcn_wmma_f32_16x16x4_f32(false, a, false, b3, (short)0, acc3, false, false);
  }

  float* crow = C + (size_t)(rowBase + half * 8) * ncol + colBase + l15;
#pragma unroll
  for (int r = 0; r < 8; ++r) {
    float* cr = crow + (size_t)r * ncol;
    cr[0]  = acc0[r];
    cr[16] = acc1[r];
    cr[32] = acc2[r];
    cr[48] = acc3[r];
  }
}

// ---------------------------------------------------------------------------
// el[n,h] = sum_f fs[n,h,f]*al[h,f]; er likewise. One wave per node.
// ---------------------------------------------------------------------------
__global__ void attn_coef(const float* __restrict__ fs,
                          const float* __restrict__ al,
                          const float* __restrict__ ar,
                          float* __restrict__ el, float* __restrict__ er,
                          int n, int fout) {
  const int node = (blockIdx.x * blockDim.x + threadIdx.x) >> 5;
  const int lane = threadIdx.x & 31;
  if (node >= n) return;
  const float* base = fs + (size_t)node * HEADS * fout;
#pragma unroll
  for (int h = 0; h < HEADS; ++h) {
    float a = 0.f, b = 0.f;
    for (int f = lane; f < fout; f += 32) {
      float v = base[h * fout + f];
      a += v * al[h * fout + f];
      b += v * ar[h * fout + f];
    }
#pragma unroll
    for (int o = 16; o; o >>= 1) {
      a += __shfl_xor(a, o);
      b += __shfl_xor(b, o);
    }
    if (lane == 0) {
      el[node * HEADS + h] = a;
      er[node * HEADS + h] = b;
    }
  }
}

__global__ void fill_f32(float* __restrict__ p, int n, float v) {
  int t = blockIdx.x * blockDim.x + threadIdx.x;
  if (t < n) p[t] = v;
}

// Monotone-toward-max float atomic max (mem initialized to -inf).
__device__ inline void atomicMaxF(float* addr, float v) {
  if (v >= 0.f) atomicMax((int*)addr, __float_as_int(v));
  else          atomicMin((unsigned int*)addr, __float_as_uint(v));
}

__device__ inline float lrelu(float x) { return x > 0.f ? x : NEG_SLOPE * x; }

// ---------------------------------------------------------------------------
// Pass 1: e = leaky(el[src]+er[dst]); store; atomic segment-max into m[dst].
// ---------------------------------------------------------------------------
__global__ void edge_logits(const int* __restrict__ src, const int* __restrict__ dst,
                            const float* __restrict__ el, const float* __restrict__ er,
                            float* __restrict__ eB, float* __restrict__ mB, int E) {
  int t = blockIdx.x * blockDim.x + threadIdx.x;
  if (t >= E) return;
  int s = src[t], d = dst[t];
  float4 a = *(const float4*)(el + (size_t)s * HEADS);
  float4 b = *(const float4*)(er + (size_t)d * HEADS);
  float4 e;
  e.x = lrelu(a.x + b.x);
  e.y = lrelu(a.y + b.y);
  e.z = lrelu(a.z + b.z);
  e.w = lrelu(a.w + b.w);
  *(float4*)(eB + (size_t)t * HEADS) = e;
  float* m = mB + (size_t)d * HEADS;
  atomicMaxF(m + 0, e.x);
  atomicMaxF(m + 1, e.y);
  atomicMaxF(m + 2, e.z);
  atomicMaxF(m + 3, e.w);
}

// ---------------------------------------------------------------------------
// Pass 2: ex = exp(e - m[dst]) stored back in place; atomic segment-sum s[dst].
// ---------------------------------------------------------------------------
__global__ void edge_exp(const int* __restrict__ dst,
                         const float* __restrict__ mB,
                         float* __restrict__ eB, float* __restrict__ sB, int E) {
  int t = blockIdx.x * blockDim.x + threadIdx.x;
  if (t >= E) return;
  int d = dst[t];
  float4 e = *(const float4*)(eB + (size_t)t * HEADS);
  float4 m = *(const float4*)(mB + (size_t)d * HEADS);
  float4 x;
  x.x = expf(e.x - m.x);
  x.y = expf(e.y - m.y);
  x.z = expf(e.z - m.z);
  x.w = expf(e.w - m.w);
  *(float4*)(eB + (size_t)t * HEADS) = x;
  float* s = sB + (size_t)d * HEADS;
  atomicAdd(s + 0, x.x);
  atomicAdd(s + 1, x.y);
  atomicAdd(s + 2, x.z);
  atomicAdd(s + 3, x.w);
}

// ---------------------------------------------------------------------------
// Pass 3: agg[dst,h,:] += fs[src,h,:] * (ex/s[dst,h]). One wave per edge;
// 32-lane strided chunks are head-uniform (fout is a multiple of 32).
// This is the HBM-roofline phase: prefetch the gather stream.
// ---------------------------------------------------------------------------
__global__ void edge_aggregate(const int* __restrict__ src, const int* __restrict__ dst,
                               const float* __restrict__ eB, const float* __restrict__ sB,
                               const float* __restrict__ fs, float* __restrict__ agg,
                               int E, int fout, int foutLog) {
  const int edge = (blockIdx.x * blockDim.x + threadIdx.x) >> 5;
  const int lane = threadIdx.x & 31;
  if (edge >= E) return;
  const int s = src[edge], d = dst[edge];
  float4 ex = *(const float4*)(eB + (size_t)edge * HEADS);
  float4 sv = *(const float4*)(sB + (size_t)d * HEADS);
  const float w0 = ex.x / sv.x, w1 = ex.y / sv.y, w2 = ex.z / sv.z, w3 = ex.w / sv.w;
  const float* fp = fs  + (size_t)s * (fout << 2);
  float*       ap = agg + (size_t)d * (fout << 2);
  const int total = fout << 2;
  // Kick the gather stream into GL2 ahead of use (global_prefetch_b8).
  __builtin_prefetch(fp + lane * 32, 0, 1);
  for (int base = 0; base < total; base += 32) {
    const int h = base >> foutLog;                       // uniform per chunk
    const float w = (h == 0) ? w0 : (h == 1) ? w1 : (h == 2) ? w2 : w3;
    const int i = base + lane;
    atomicAdd(ap + i, fp[i] * w);
  }
}

// ---------------------------------------------------------------------------
// Head-mean + bias (+ LayerNorm + ReLU). One wave per node, PER = fout/32.
// ---------------------------------------------------------------------------
template <int PER>
__global__ void pool_norm(const float* __restrict__ agg, const float* __restrict__ bias,
                          const float* __restrict__ g, const float* __restrict__ beta,
                          float* __restrict__ out, int n, int fout, int doNorm) {
  const int node = (blockIdx.x * blockDim.x + threadIdx.x) >> 5;
  const int lane = threadIdx.x & 31;
  if (node >= n) return;
  const float* a = agg + (size_t)node * HEADS * fout;
  float v[PER];
  float sum = 0.f;
#pragma unroll
  for (int j = 0; j < PER; ++j) {
    const int f = j * 32 + lane;
    float t = 0.f;
#pragma unroll
    for (int h = 0; h < HEADS; ++h) t += a[h * fout + f] + bias[h * fout + f];
    t *= 0.25f;
    v[j] = t;
    sum += t;
  }
#pragma unroll
  for (int o = 16; o; o >>= 1) sum += __shfl_xor(sum, o);
  if (doNorm) {
    const float mu = sum / (float)fout;
    float var = 0.f;
#pragma unroll
    for (int j = 0; j < PER; ++j) { float dd = v[j] - mu; var += dd * dd; }
#pragma unroll
    for (int o = 16; o; o >>= 1) var += __shfl_xor(var, o);
    var /= (float)fout;
    const float inv = rsqrtf(var + 1e-5f);
#pragma unroll
    for (int j = 0; j < PER; ++j) {
      const int f = j * 32 + lane;
      float x = (v[j] - mu) * inv * g[f] + beta[f];
      out[(size_t)node * fout + f] = x > 0.f ? x : 0.f;
    }
  } else {
#pragma unroll
    for (int j = 0; j < PER; ++j)
      out[(size_t)node * fout + j * 32 + lane] = v[j];
  }
}

// ---------------------------------------------------------------------------
extern "C" void kernel_launch(void* const* d_in, const int* in_sizes, int n_in,
                              void* d_out, int out_size, void* d_ws, size_t ws_size,
                              hipStream_t stream) {
  (void)n_in; (void)out_size; (void)ws_size;
  const float* feat = (const float*)d_in[0];
  const int*   src  = (const int*)d_in[1];
  const int*   dst  = (const int*)d_in[2];
  const int N = in_sizes[0] / 128;
  const int E = in_sizes[1];

  // setup_inputs() dict order: W0,al0,ar0,b0,g0,beta0, W1,al1,ar1,b1,g1,beta1, W2,al2,ar2,b2
  const float* W[3]  = {(const float*)d_in[3],  (const float*)d_in[9],  (const float*)d_in[15]};
  const float* al[3] = {(const float*)d_in[4],  (const float*)d_in[10], (const float*)d_in[16]};
  const float* ar[3] = {(const float*)d_in[5],  (const float*)d_in[11], (const float*)d_in[17]};
  const float* bb[3] = {(const float*)d_in[6],  (const float*)d_in[12], (const float*)d_in[18]};
  const float* gg[2] = {(const float*)d_in[7],  (const float*)d_in[13]};
  const float* be[2] = {(const float*)d_in[8],  (const float*)d_in[14]};

  char* ws = (char*)d_ws;
  size_t off = 0;
  auto carve = [&](size_t bytes) -> void* {
    void* p = ws + off;
    off += (bytes + 255) & ~(size_t)255;
    return p;
  };
  float* fsB  = (float*)carve((size_t)N * 1024 * sizeof(float));  // [N, H*fout] (max 1024)
  float* aggB = (float*)carve((size_t)N * 1024 * sizeof(float));  // [N, H*fout]
  float* hA   = (float*)carve((size_t)N * 256 * sizeof(float));   // inter-layer features
  float* hB   = (float*)carve((size_t)N * 256 * sizeof(float));
  float* elB  = (float*)carve((size_t)N * HEADS * sizeof(float));
  float* erB  = (float*)carve((size_t)N * HEADS * sizeof(float));
  float* mB   = (float*)carve((size_t)N * HEADS * sizeof(float));
  float* sB   = (float*)carve((size_t)N * HEADS * sizeof(float));
  float* eB   = (float*)carve((size_t)E * HEADS * sizeof(float)); // logits then exp, in place

  const int fins[3]  = {128, 256, 256};
  const int fouts[3] = {256, 256, 64};
  const float* ins[3] = {feat, hA, hB};
  float* outs[3] = {hA, hB, (float*)d_out};

  const int nodeWaveBlocks = (N + 7) / 8;   // 256 threads = 8 waves
  const int edgeBlocks     = (E + 255) / 256;
  const int edgeWaveBlocks = (E + 7) / 8;

  for (int L = 0; L < 3; ++L) {
    const int fin = fins[L], fout = fouts[L], ncol = HEADS * fout;
    // 1) fs = h @ W  (WMMA f32; each block covers 16 rows x 256 cols)
    dim3 grid(N / 16, ncol / 256);
    gemm_wmma_f32<<<grid, 128, 0, stream>>>(ins[L], W[L], fsB, fin, ncol);
    // 2) attention coefficients
    attn_coef<<<nodeWaveBlocks, 256, 0, stream>>>(fsB, al[L], ar[L], elB, erB, N, fout);
    // 3) init segment stats + accumulator
    fill_f32<<<(N * HEADS + 255) / 256, 256, 0, stream>>>(mB, N * HEADS, -INFINITY);
    hipMemsetAsync(sB, 0, (size_t)N * HEADS * sizeof(float), stream);
    hipMemsetAsync(aggB, 0, (size_t)N * ncol * sizeof(float), stream);
    // 4) edge softmax
    edge_logits<<<edgeBlocks, 256, 0, stream>>>(src, dst, elB, erB, eB, mB, E);
    edge_exp<<<edgeBlocks, 256, 0, stream>>>(dst, mB, eB, sB, E);
    // 5) weighted aggregation (HBM roofline phase)
    const int foutLog = (fout == 256) ? 8 : 6;
    edge_aggregate<<<edgeWaveBlocks, 256, 0, stream>>>(src, dst, eB, sB, fsB, aggB, E, fout, foutLog);
    // 6) head-mean + bias (+ LN + ReLU)
    if (L < 2)
      pool_norm<8><<<nodeWaveBlocks, 256, 0, stream>>>(aggB, bb[L], gg[L], be[L], outs[L], N, fout, 1);
    else
      pool_norm<2><<<nodeWaveBlocks, 256, 0, stream>>>(aggB, bb[L], nullptr, nullptr, outs[L], N, fout, 0);
  }
}